// BiCircuitGNN_66571993088626
// MI455X (gfx1250) — compile-verified
//
#include <hip/hip_runtime.h>

#define HID 128
#define NQ 32
#define NND 65536
#define NED 524288
#define NGR 64
#define BN_EPS 1e-5f

typedef __attribute__((ext_vector_type(2))) float v2f;
typedef __attribute__((ext_vector_type(8))) float v8f;

// ---------------------------------------------------------------------------
// GEMM: Y[rows x 128] = op(X [+ X2]) @ W[K x 128] + Bias, optional ReLU.
// One wave (32 threads) per 16-row block; 8 column tiles of 16 via
// V_WMMA_F32_16X16X4_F32 (full fp32 path — workload is HBM-bound, so fp32 is free).
// Fragment layouts per CDNA5 ISA 7.12.2 (32-bit A 16x4, B 4x16, C/D 16x16).
// ---------------------------------------------------------------------------
template<int K, bool ADD2, bool RELU>
__global__ __launch_bounds__(32)
void wmma_gemm_rowsx128(const float* __restrict__ X,
                        const float* __restrict__ X2,
                        const float* __restrict__ W,
                        const float* __restrict__ Bias,
                        float* __restrict__ Y)
{
    const int lane = threadIdx.x;        // 0..31
    const int half = lane >> 4;          // 0 | 1
    const int m    = lane & 15;
    const size_t r0 = (size_t)blockIdx.x * 16;

    const float* xr  = X + (r0 + m) * K;
    const float* x2r = ADD2 ? (X2 + (r0 + m) * K) : X;

    v8f acc[8];
#pragma unroll
    for (int t = 0; t < 8; ++t)
#pragma unroll
        for (int v = 0; v < 8; ++v) acc[t][v] = 0.0f;

    for (int k = 0; k < K; k += 4) {
        const int ka = k + 2 * half;     // VGPR0 holds K=k (lanes0-15) / K=k+2 (lanes16-31)
        float a0 = xr[ka];
        float a1 = xr[ka + 1];
        if (ADD2) { a0 += x2r[ka]; a1 += x2r[ka + 1]; }
        v2f a; a.x = a0; a.y = a1;

        const float* wp = W + (size_t)ka * HID + m;
#pragma unroll
        for (int t = 0; t < 8; ++t) {
            v2f b;
            b.x = wp[t * 16];            // B row ka,   col t*16+m
            b.y = wp[HID + t * 16];      // B row ka+1, col t*16+m
            acc[t] = __builtin_amdgcn_wmma_f32_16x16x4_f32(
                false, a, false, b, (short)0, acc[t], false, false);
        }
    }

#pragma unroll
    for (int t = 0; t < 8; ++t) {
        const int col = t * 16 + m;
        const float bias = Bias[col];
#pragma unroll
        for (int v = 0; v < 8; ++v) {
            float val = acc[t][v] + bias;
            if (RELU) val = fmaxf(val, 0.0f);
            Y[(r0 + (size_t)(v + 8 * half)) * HID + col] = val;  // C/D: row v+8*half, col lane
        }
    }
}

// ---------------------------------------------------------------------------
// Elementwise / scatter kernels
// ---------------------------------------------------------------------------
__global__ void zero_f(float* __restrict__ p, size_t n)
{
    size_t i = (size_t)blockIdx.x * blockDim.x + threadIdx.x;
    const size_t stride = (size_t)gridDim.x * blockDim.x;
    for (; i < n; i += stride) p[i] = 0.0f;
}

// msg = relu(h[src] + ea); agg[dst] += msg.  32 lanes/edge, float4/lane.
__global__ __launch_bounds__(256)
void msg_scatter(const float* __restrict__ H, const float* __restrict__ EA,
                 const int* __restrict__ src, const int* __restrict__ dst,
                 float* __restrict__ AGG)
{
    const size_t idx = (size_t)blockIdx.x * 256 + threadIdx.x;   // NED*32 threads total
    const int e = (int)(idx >> 5);
    const int q = ((int)idx & 31) * 4;
    const int s = src[e];
    const int d = dst[e];
    const float4 hv = *(const float4*)(H + (size_t)s * HID + q);
    const float4 ev = *(const float4*)(EA + (size_t)e * HID + q);
    float4 msg;
    msg.x = fmaxf(hv.x + ev.x, 0.0f);
    msg.y = fmaxf(hv.y + ev.y, 0.0f);
    msg.z = fmaxf(hv.z + ev.z, 0.0f);
    msg.w = fmaxf(hv.w + ev.w, 0.0f);
    float* out = AGG + (size_t)d * HID + q;
    atomicAdd(out + 0, msg.x);
    atomicAdd(out + 1, msg.y);
    atomicAdd(out + 2, msg.z);
    atomicAdd(out + 3, msg.w);
}

// Per-column sum / sum-of-squares over the node dimension (coalesced: thread=col).
__global__ __launch_bounds__(128)
void bn_stats(const float* __restrict__ X, float* __restrict__ sums,
              float* __restrict__ sumsq)
{
    const int col = threadIdx.x;                 // 128 threads
    const size_t row0 = (size_t)blockIdx.x * 256;
    float s = 0.0f, s2 = 0.0f;
    for (int r = 0; r < 256; ++r) {
        const float v = X[(row0 + r) * HID + col];
        s += v;
        s2 += v * v;
    }
    atomicAdd(&sums[col], s);
    atomicAdd(&sumsq[col], s2);
}

__global__ __launch_bounds__(128)
void bn_finalize(const float* __restrict__ sums, const float* __restrict__ sumsq,
                 const float* __restrict__ g, const float* __restrict__ b,
                 float* __restrict__ scale, float* __restrict__ shift)
{
    const int c = threadIdx.x;
    const float inv_n = 1.0f / (float)NND;
    const float mean = sums[c] * inv_n;
    const float var  = sumsq[c] * inv_n - mean * mean;
    const float sc   = g[c] * rsqrtf(var + BN_EPS);
    scale[c] = sc;
    shift[c] = b[c] - mean * sc;
}

__global__ __launch_bounds__(256)
void bn_apply_relu(const float* __restrict__ X, const float* __restrict__ scale,
                   const float* __restrict__ shift, float* __restrict__ Y)
{
    const size_t idx = (size_t)blockIdx.x * 256 + threadIdx.x;   // one float4 each
    const int c = ((int)idx & 31) * 4;
    const float4 v = *(const float4*)(X + idx * 4);
    float4 r;
    r.x = fmaxf(v.x * scale[c + 0] + shift[c + 0], 0.0f);
    r.y = fmaxf(v.y * scale[c + 1] + shift[c + 1], 0.0f);
    r.z = fmaxf(v.z * scale[c + 2] + shift[c + 2], 0.0f);
    r.w = fmaxf(v.w * scale[c + 3] + shift[c + 3], 0.0f);
    *(float4*)(Y + idx * 4) = r;
}

// pooled[batch[n]][0:128]=sum xf, [128:256]=sum xb
__global__ __launch_bounds__(256)
void pool_kernel(const float* __restrict__ XF, const float* __restrict__ XB,
                 const int* __restrict__ batch, float* __restrict__ P)
{
    const size_t idx = (size_t)blockIdx.x * 256 + threadIdx.x;   // NND*64 threads
    const int n = (int)(idx >> 6);
    const int c = ((int)idx & 63) * 4;
    const float4 v = (c < HID)
        ? *(const float4*)(XF + (size_t)n * HID + c)
        : *(const float4*)(XB + (size_t)n * HID + (c - HID));
    float* out = P + (size_t)batch[n] * (2 * HID) + c;
    atomicAdd(out + 0, v.x);
    atomicAdd(out + 1, v.y);
    atomicAdd(out + 2, v.z);
    atomicAdd(out + 3, v.w);
}

// out[g] = relu(pooled[g] @ W1 + b1) @ W2 + b2   (tiny: one block per graph)
__global__ __launch_bounds__(128)
void head_kernel(const float* __restrict__ P, const float* __restrict__ W1,
                 const float* __restrict__ B1, const float* __restrict__ W2,
                 const float* __restrict__ B2, float* __restrict__ out)
{
    __shared__ float red[HID];
    const int g = blockIdx.x;
    const int t = threadIdx.x;                   // 128
    float acc = B1[t];
    for (int k = 0; k < 2 * HID; ++k)
        acc += P[(size_t)g * (2 * HID) + k] * W1[(size_t)k * HID + t];
    acc = fmaxf(acc, 0.0f);
    red[t] = acc * W2[t];
    __syncthreads();
    for (int s = 64; s > 0; s >>= 1) {
        if (t < s) red[t] += red[t + s];
        __syncthreads();
    }
    if (t == 0) out[g] = red[0] + B2[0];
}

// ---------------------------------------------------------------------------
extern "C" void kernel_launch(void* const* d_in, const int* in_sizes, int n_in,
                              void* d_out, int out_size, void* d_ws, size_t ws_size,
                              hipStream_t stream)
{
    const float* x      = (const float*)d_in[0];
    const float* eattr  = (const float*)d_in[1];
    const int*   eindex = (const int*)d_in[2];
    const int*   batch  = (const int*)d_in[3];
    const float* node_w = (const float*)d_in[4];
    const float* node_b = (const float*)d_in[5];
    const float* edge_w = (const float*)d_in[6];
    const float* edge_b = (const float*)d_in[7];
    const int* src = eindex;
    const int* dst = eindex + NED;

    float* ws = (float*)d_ws;
    const size_t N128 = (size_t)NND * HID;
    float* h      = ws;
    float* agg    = ws + 1 * N128;
    float* bufA   = ws + 2 * N128;
    float* bufB   = ws + 3 * N128;
    float* t1     = ws + 4 * N128;
    float* xf     = ws + 5 * N128;
    float* xb     = ws + 6 * N128;
    float* ea     = ws + 7 * N128;               // [NED x 128]
    float* sums   = ea + (size_t)NED * HID;
    float* sumsq  = sums + HID;
    float* scale  = sumsq + HID;
    float* shift  = scale + HID;
    float* pooled = shift + HID;                 // [64 x 256]

    // Input linears (WMMA fp32)
    wmma_gemm_rowsx128<2 * NQ, false, false><<<NND / 16, 32, 0, stream>>>(
        x, x, node_w, node_b, h);
    wmma_gemm_rowsx128<NQ, false, false><<<NED / 16, 32, 0, stream>>>(
        eattr, eattr, edge_w, edge_b, ea);

    auto gine_bn = [&](const float* Hin, const int* sA, const int* dA,
                       const float* w1, const float* b1,
                       const float* w2, const float* b2,
                       const float* bng, const float* bnb, float* Hout) {
        zero_f<<<4096, 256, 0, stream>>>(agg, N128);
        msg_scatter<<<(NED * 32) / 256, 256, 0, stream>>>(Hin, ea, sA, dA, agg);
        // t1 = relu((Hin + agg) @ w1 + b1)
        wmma_gemm_rowsx128<HID, true, true><<<NND / 16, 32, 0, stream>>>(
            Hin, agg, w1, b1, t1);
        // bufB = t1 @ w2 + b2
        wmma_gemm_rowsx128<HID, false, false><<<NND / 16, 32, 0, stream>>>(
            t1, t1, w2, b2, bufB);
        zero_f<<<1, 256, 0, stream>>>(sums, 2 * HID);            // sums + sumsq
        bn_stats<<<NND / 256, HID, 0, stream>>>(bufB, sums, sumsq);
        bn_finalize<<<1, HID, 0, stream>>>(sums, sumsq, bng, bnb, scale, shift);
        bn_apply_relu<<<(N128 / 4) / 256, 256, 0, stream>>>(bufB, scale, shift, Hout);
    };

    // forward flow (src -> dst)
    gine_bn(h, src, dst,
            (const float*)d_in[8], (const float*)d_in[9],
            (const float*)d_in[10], (const float*)d_in[11],
            (const float*)d_in[24], (const float*)d_in[25], bufA);
    gine_bn(bufA, src, dst,
            (const float*)d_in[12], (const float*)d_in[13],
            (const float*)d_in[14], (const float*)d_in[15],
            (const float*)d_in[26], (const float*)d_in[27], xf);
    // backward flow (dst -> src)
    gine_bn(h, dst, src,
            (const float*)d_in[16], (const float*)d_in[17],
            (const float*)d_in[18], (const float*)d_in[19],
            (const float*)d_in[28], (const float*)d_in[29], bufA);
    gine_bn(bufA, dst, src,
            (const float*)d_in[20], (const float*)d_in[21],
            (const float*)d_in[22], (const float*)d_in[23],
            (const float*)d_in[30], (const float*)d_in[31], xb);

    // pooling + head
    zero_f<<<64, 256, 0, stream>>>(pooled, (size_t)NGR * 2 * HID);
    pool_kernel<<<((size_t)NND * 64) / 256, 256, 0, stream>>>(xf, xb, batch, pooled);
    head_kernel<<<NGR, HID, 0, stream>>>(
        pooled, (const float*)d_in[32], (const float*)d_in[33],
        (const float*)d_in[34], (const float*)d_in[35], (float*)d_out);
}